// EffiGlobalPointer_11020886082004
// MI455X (gfx1250) — compile-verified
//
#include <hip/hip_runtime.h>
#include <hip/hip_bf16.h>

typedef __attribute__((ext_vector_type(2))) float v2f;
typedef __attribute__((ext_vector_type(8))) float v8f;

#define NEGV (-1000000000000.0f)

constexpr int B_ = 8, S_ = 1024, H_ = 768, F_ = 32, HID_ = 800;
constexpr int N1_ = 128, N2_ = 18, D_ = 64, T_ = 9;
constexpr int ROWS_ = B_ * S_; // 8192

// ---------------------------------------------------------------------------
// Kernel 1: fused GEMM1 (h@W1+b1 -> out1) and GEMM2 (h@W2+b2 -> biasT/2)
// One wave per 16x16 output tile; K=800 -> 200 x V_WMMA_F32_16X16X4_F32.
// K-loop split at 768 so the lhs/hand concat needs no per-iteration address
// selects; W2 guard is done on DATA (mul by 0/1) with a clamped index so no
// exec-mask branches appear around the loads.
// ---------------------------------------------------------------------------
__global__ __launch_bounds__(256) void effigp_gemm1_kernel(
    const float* __restrict__ lhs, const float* __restrict__ hand,
    const float* __restrict__ W1, const float* __restrict__ b1,
    const float* __restrict__ W2, const float* __restrict__ b2,
    float* __restrict__ out1, float* __restrict__ biasT)
{
  const int lane = threadIdx.x & 31;
  const int wid  = blockIdx.x * 8 + (threadIdx.x >> 5); // 5120 waves
  const int mtile = wid / 10;
  const int nt    = wid % 10;                           // 0..7 -> W1, 8..9 -> W2
  const int mb = mtile * 16;
  const int lo = lane & 15;
  const int hi = lane >> 4;
  const float* arow_l = lhs  + (size_t)(mb + lo) * H_ + (hi << 1);
  const float* arow_h = hand + (size_t)(mb + lo) * F_ + (hi << 1);

  v8f c = {};
  if (nt < 8) {
    const int n = nt * 16 + lo;
    const float* wcol = W1 + n;
#pragma unroll 4
    for (int kk = 0; kk < H_; kk += 4) {                // 192 iters: lhs half
      const int k0 = kk + (hi << 1);
      v2f a = *(const v2f*)(arow_l + kk);
      v2f bb;
      bb.x = wcol[(size_t)k0 * N1_];
      bb.y = wcol[(size_t)(k0 + 1) * N1_];
      c = __builtin_amdgcn_wmma_f32_16x16x4_f32(false, a, false, bb,
                                                (short)0, c, false, false);
    }
#pragma unroll
    for (int kk = H_; kk < HID_; kk += 4) {             // 8 iters: hand half
      const int k0 = kk + (hi << 1);
      v2f a = *(const v2f*)(arow_h + (kk - H_));
      v2f bb;
      bb.x = wcol[(size_t)k0 * N1_];
      bb.y = wcol[(size_t)(k0 + 1) * N1_];
      c = __builtin_amdgcn_wmma_f32_16x16x4_f32(false, a, false, bb,
                                                (short)0, c, false, false);
    }
    const float bias = b1[n];
#pragma unroll
    for (int i = 0; i < 8; ++i) {
      const int r = mb + i + (hi << 3);
      out1[(size_t)r * N1_ + n] = c[i] + bias;
    }
  } else {
    const int n = (nt - 8) * 16 + lo;
    const float vf = (n < N2_) ? 1.0f : 0.0f;           // data-side guard
    const int nc = (n < N2_) ? n : 0;                   // always-valid address
    const float* wcol = W2 + nc;
#pragma unroll 4
    for (int kk = 0; kk < H_; kk += 4) {
      const int k0 = kk + (hi << 1);
      v2f a = *(const v2f*)(arow_l + kk);
      v2f bb;
      bb.x = wcol[(size_t)k0 * N2_] * vf;
      bb.y = wcol[(size_t)(k0 + 1) * N2_] * vf;
      c = __builtin_amdgcn_wmma_f32_16x16x4_f32(false, a, false, bb,
                                                (short)0, c, false, false);
    }
#pragma unroll
    for (int kk = H_; kk < HID_; kk += 4) {
      const int k0 = kk + (hi << 1);
      v2f a = *(const v2f*)(arow_h + (kk - H_));
      v2f bb;
      bb.x = wcol[(size_t)k0 * N2_] * vf;
      bb.y = wcol[(size_t)(k0 + 1) * N2_] * vf;
      c = __builtin_amdgcn_wmma_f32_16x16x4_f32(false, a, false, bb,
                                                (short)0, c, false, false);
    }
    if (n < N2_) {
      const float bias = b2[n];
#pragma unroll
      for (int i = 0; i < 8; ++i) {
        const int r = mb + i + (hi << 3);
        const int bidx = r >> 10, sidx = r & (S_ - 1);
        // transposed, halved: biasT[b][n][s] = (h@W2+b2)/2
        biasT[((size_t)bidx * N2_ + n) * S_ + sidx] = (c[i] + bias) * 0.5f;
      }
    }
  }
}

// ---------------------------------------------------------------------------
// Kernel 2: deinterleave out1 -> qw/kw and apply RoPE. One thread per (row,d).
// qw[d]=out[2d], kw[d]=out[2d+1]; rotation pairs (2j,2j+1) of qw/kw map to
// out columns (4j,4j+2) and (4j+1,4j+3).
// ---------------------------------------------------------------------------
__global__ __launch_bounds__(256) void effigp_rope_kernel(
    const float* __restrict__ out1, float* __restrict__ qw, float* __restrict__ kw)
{
  const int idx = blockIdx.x * 256 + threadIdx.x;       // r*64 + d
  const int r = idx >> 6;
  const int d = idx & 63;
  const int j = d >> 1;
  const float pos  = (float)(r & (S_ - 1));
  const float freq = powf(10000.0f, -2.0f * (float)j / (float)D_);
  const float ang  = pos * freq;
  const float cs = cosf(ang), sn = sinf(ang);
  const float* o = out1 + (size_t)r * N1_;
  float q, k;
  if ((d & 1) == 0) {
    q = o[4 * j]     * cs - o[4 * j + 2] * sn;
    k = o[4 * j + 1] * cs - o[4 * j + 3] * sn;
  } else {
    q = o[4 * j + 2] * cs + o[4 * j]     * sn;
    k = o[4 * j + 3] * cs + o[4 * j + 1] * sn;
  }
  qw[idx] = q;
  kw[idx] = k;
}

// ---------------------------------------------------------------------------
// Kernel 3: logits. One wave per 16x32 (m,n) tile: 2 WMMA accumulators over
// K=64 (32x v_wmma_f32_16x16x4_f32), then 9 bias planes with mask+tril folded
// into one FMA each, streamed out with non-temporal stores (302 MB > 192 MB
// L2 -> don't cache). The two 64B half-lines of each 128B line come from the
// same wave back-to-back for write combining.
// ---------------------------------------------------------------------------
__global__ __launch_bounds__(256) void effigp_logits_kernel(
    const float* __restrict__ qw, const float* __restrict__ kw,
    const float* __restrict__ biasT, const int* __restrict__ am,
    float* __restrict__ out)
{
  const int lane = threadIdx.x & 31;
  const int wid  = blockIdx.x * 8 + (threadIdx.x >> 5); // 16384 waves
  const int b    = wid >> 11;                           // /(64*32)
  const int rem  = wid & 2047;
  const int mt = rem >> 5, ntl = rem & 31;
  const int mb = mt * 16, nb = ntl * 32;
  const int lo = lane & 15, hi = lane >> 4;

  const float* qrow  = qw + ((size_t)(b * S_ + mb + lo)) * D_ + (hi << 1);
  const float* krow0 = kw + ((size_t)(b * S_ + nb + lo)) * D_ + (hi << 1);
  const float* krow1 = krow0 + (size_t)16 * D_;

  v8f c0 = {}, c1 = {};
#pragma unroll
  for (int kk = 0; kk < D_; kk += 4) {
    v2f a  = *(const v2f*)(qrow + kk);    // A: qw rows (M x K)
    v2f b0 = *(const v2f*)(krow0 + kk);   // B: kw^T    (K x N), cols nb..nb+15
    v2f b1v = *(const v2f*)(krow1 + kk);  //            cols nb+16..nb+31
    c0 = __builtin_amdgcn_wmma_f32_16x16x4_f32(false, a, false, b0,
                                               (short)0, c0, false, false);
    c1 = __builtin_amdgcn_wmma_f32_16x16x4_f32(false, a, false, b1v,
                                               (short)0, c1, false, false);
  }

  const int n0 = nb + lo;
  const int n1 = n0 + 16;
  const float mk0 = (float)am[b * S_ + n0];
  const float mk1 = (float)am[b * S_ + n1];
  float ms0[8], ma0[8], ms1[8], ma1[8];
#pragma unroll
  for (int i = 0; i < 8; ++i) {
    const int m = mb + i + (hi << 3);
    const float mq = (float)am[b * S_ + m];
    // ((x*mq + NEG(1-mq))*mk + NEG(1-mk)) + tril*NEG  ==  x*(mq*mk) + Madd
    ms0[i] = mq * mk0;
    ma0[i] = NEGV * (1.0f - mq) * mk0 + NEGV * (1.0f - mk0)
           + ((m > n0) ? NEGV : 0.0f);
    ms1[i] = mq * mk1;
    ma1[i] = NEGV * (1.0f - mq) * mk1 + NEGV * (1.0f - mk1)
           + ((m > n1) ? NEGV : 0.0f);
    c0[i] *= 0.125f;                      // 1/sqrt(64)
    c1[i] *= 0.125f;
  }

  const float* bT = biasT + (size_t)b * N2_ * S_;
  float* op = out + (((size_t)(b * T_) * S_ + mb) * S_ + n0);
#pragma unroll
  for (int t = 0; t < T_; ++t) {
    const float* bqt = bT + (size_t)(2 * t) * S_;
    const float* bkt = bT + (size_t)(2 * t + 1) * S_;
    const float bk0 = bkt[n0];
    const float bk1 = bkt[n1];
#pragma unroll
    for (int i = 0; i < 8; ++i) {
      const int m = mb + i + (hi << 3);
      const float bq = bqt[m];
      float* rp = op + (size_t)t * S_ * S_ + (size_t)(i + (hi << 3)) * S_;
      __builtin_nontemporal_store((c0[i] + bq + bk0) * ms0[i] + ma0[i], rp);
      __builtin_nontemporal_store((c1[i] + bq + bk1) * ms1[i] + ma1[i], rp + 16);
    }
  }
}

// ---------------------------------------------------------------------------
extern "C" void kernel_launch(void* const* d_in, const int* in_sizes, int n_in,
                              void* d_out, int out_size, void* d_ws, size_t ws_size,
                              hipStream_t stream) {
  const float* lhs  = (const float*)d_in[0]; // (8,1024,768)
  const float* hand = (const float*)d_in[1]; // (8,1024,32)
  const int*   am   = (const int*)  d_in[2]; // (8,1024)
  const float* W1   = (const float*)d_in[3]; // (800,128)
  const float* b1   = (const float*)d_in[4]; // (128,)
  const float* W2   = (const float*)d_in[5]; // (800,18)
  const float* b2   = (const float*)d_in[6]; // (18,)
  float* out = (float*)d_out;                // (8,9,1024,1024)

  float* ws    = (float*)d_ws;
  float* out1  = ws;                                   // 8192*128
  float* qw    = out1 + (size_t)ROWS_ * N1_;           // 8192*64
  float* kw    = qw   + (size_t)ROWS_ * D_;            // 8192*64
  float* biasT = kw   + (size_t)ROWS_ * D_;            // 8*18*1024

  // 512 m-tiles * 10 n-tiles = 5120 waves -> 640 blocks of 8 waves
  effigp_gemm1_kernel<<<640, 256, 0, stream>>>(lhs, hand, W1, b1, W2, b2,
                                               out1, biasT);
  // 8192*64 threads
  effigp_rope_kernel<<<(ROWS_ * D_) / 256, 256, 0, stream>>>(out1, qw, kw);
  // 8*64*32 = 16384 waves -> 2048 blocks
  effigp_logits_kernel<<<2048, 256, 0, stream>>>(qw, kw, biasT, am, out);
}